// GNN_node_10153302688344
// MI455X (gfx1250) — compile-verified
//
#include <hip/hip_runtime.h>
#include <hip/hip_bf16.h>
#include <float.h>
#include <stdint.h>

// ---------------------------------------------------------------------------
// Problem constants (match reference)
// ---------------------------------------------------------------------------
#define N_NODES 300000
#define N_NETS  200000
#define E_SINK  1000000
#define NUM_VN  1000
#define EMB     64
#define SLOPE   0.01f

typedef __attribute__((ext_vector_type(2))) float v2f;
typedef __attribute__((ext_vector_type(8))) float v8f;

__device__ __forceinline__ v8f wmma4(v2f a, v2f b, v8f c) {
  // D = A(16x4, f32) * B(4x16, f32) + C(16x16, f32)
  return __builtin_amdgcn_wmma_f32_16x16x4_f32(
      /*neg_a=*/false, a, /*neg_b=*/false, b,
      /*c_mod=*/(short)0, c, /*reuse_a=*/false, /*reuse_b=*/false);
}

__device__ __forceinline__ float leaky(float x) {
  return x >= 0.0f ? x : SLOPE * x;
}

__device__ __forceinline__ void atomicMaxF(float* addr, float val) {
  int* ia = (int*)addr;
  int old = __float_as_int(*addr);
  while (__int_as_float(old) < val) {
    int assumed = old;
    old = atomicCAS(ia, assumed, __float_as_int(val));
    if (old == assumed) break;
  }
}

// ---------------------------------------------------------------------------
// CDNA5 async global->LDS weight staging.
// Generic pointers to LDS carry the LDS byte offset in their low 32 bits
// (ISA 10.2); async-to-LDS wants that offset in VDST.
// ---------------------------------------------------------------------------
__device__ __forceinline__ void async_stage_issue(float* lds_dst, const float* gsrc,
                                                  int n_floats, int tid, int nthr) {
  for (int i = tid * 4; i < n_floats; i += nthr * 4) {
    unsigned loff = (unsigned)(uintptr_t)(lds_dst + i);
    const float* ga = gsrc + i;
    asm volatile("global_load_async_to_lds_b128 %0, %1, off"
                 :: "v"(loff), "v"(ga)
                 : "memory");
  }
}
__device__ __forceinline__ void async_stage_wait() {
  asm volatile("s_wait_asynccnt 0x0" ::: "memory");
}

#define ACC_INIT(acc, bv)                              \
  _Pragma("unroll") for (int _i = 0; _i < 8; ++_i) acc[_i] = (bv);

// Per-lane WMMA tile coordinates (wave32, 32-bit data):
//   A frag (16x4): row r = lane&15, K pair = k0 + 2*(lane>>4) .. +1
//   B frag (4x16): col c = lane&15, same K pair
//   D frag: acc[i] -> D[i + 8*(lane>>4)][lane&15]

// ---------------------------------------------------------------------------
// Node encoder: H = leaky(X@W1+b1)@W2+b2.  X 16x32, W1 32x128, W2 128x64.
// 2 waves/block; W2 staged in LDS (32KB), hidden in LDS (16.6KB).
// ---------------------------------------------------------------------------
__global__ void enc_node_kernel(const float* __restrict__ X,
                                const float* __restrict__ W1, const float* __restrict__ b1,
                                const float* __restrict__ W2, const float* __restrict__ b2,
                                float* __restrict__ H) {
  __shared__ __align__(16) float sW2[128 * 64];
  __shared__ __align__(16) float hid[2][16][130];
  int tid = threadIdx.x;
  int wave = tid >> 5, lane = tid & 31;
  int tile = blockIdx.x * 2 + wave;

  async_stage_issue(sW2, W2, 128 * 64, tid, 64);
  async_stage_wait();
  __syncthreads();
  if (tile >= N_NODES / 16) return;

  int row0 = tile * 16;
  int r = lane & 15, kh = lane >> 4, c = lane & 15;
  const float* xr = X + (size_t)(row0 + r) * 32;

  // Preload all A fragments of X (K=32 -> 8 frags), reused across 8 col tiles.
  v2f a1[8];
#pragma unroll
  for (int j = 0; j < 8; ++j) {
    int k = 4 * j + 2 * kh;
    a1[j].x = xr[k]; a1[j].y = xr[k + 1];
  }
  for (int ct = 0; ct < 8; ++ct) {
    v8f acc; float bv = b1[ct * 16 + c];
    ACC_INIT(acc, bv);
#pragma unroll
    for (int j = 0; j < 8; ++j) {
      int k = 4 * j + 2 * kh;
      v2f b; b.x = W1[(size_t)k * 128 + ct * 16 + c];
             b.y = W1[(size_t)(k + 1) * 128 + ct * 16 + c];
      acc = wmma4(a1[j], b, acc);
    }
#pragma unroll
    for (int i = 0; i < 8; ++i)
      hid[wave][i + 8 * kh][ct * 16 + c] = leaky(acc[i]);
  }
  // Preload A fragments of hidden (K=128 -> 32 frags) from LDS.
  v2f a2[32];
#pragma unroll
  for (int j = 0; j < 32; ++j) {
    int k = 4 * j + 2 * kh;
    a2[j].x = hid[wave][r][k]; a2[j].y = hid[wave][r][k + 1];
  }
  for (int ct = 0; ct < 4; ++ct) {
    v8f acc; float bv = b2[ct * 16 + c];
    ACC_INIT(acc, bv);
#pragma unroll
    for (int j = 0; j < 32; ++j) {
      int k = 4 * j + 2 * kh;
      v2f b; b.x = sW2[k * 64 + ct * 16 + c];
             b.y = sW2[(k + 1) * 64 + ct * 16 + c];
      acc = wmma4(a2[j], b, acc);
    }
#pragma unroll
    for (int i = 0; i < 8; ++i)
      H[(size_t)(row0 + i + 8 * kh) * 64 + ct * 16 + c] = acc[i];
  }
}

// ---------------------------------------------------------------------------
// Net encoder: G = leaky(X@W1+b1)@W2+b2.  X 16x16, W1 16x64, W2 64x64.
// 4 waves/block; W1+W2 staged (20KB), hidden 16.9KB.
// ---------------------------------------------------------------------------
__global__ void enc_net_kernel(const float* __restrict__ X,
                               const float* __restrict__ W1, const float* __restrict__ b1,
                               const float* __restrict__ W2, const float* __restrict__ b2,
                               float* __restrict__ G) {
  __shared__ __align__(16) float sW1[16 * 64];
  __shared__ __align__(16) float sW2[64 * 64];
  __shared__ __align__(16) float hid[4][16][66];
  int tid = threadIdx.x;
  int wave = tid >> 5, lane = tid & 31;
  int tile = blockIdx.x * 4 + wave;

  async_stage_issue(sW1, W1, 16 * 64, tid, 128);
  async_stage_issue(sW2, W2, 64 * 64, tid, 128);
  async_stage_wait();
  __syncthreads();
  if (tile >= N_NETS / 16) return;

  int row0 = tile * 16;
  int r = lane & 15, kh = lane >> 4, c = lane & 15;
  const float* xr = X + (size_t)(row0 + r) * 16;

  v2f a1[4];
#pragma unroll
  for (int j = 0; j < 4; ++j) {
    int k = 4 * j + 2 * kh;
    a1[j].x = xr[k]; a1[j].y = xr[k + 1];
  }
  for (int ct = 0; ct < 4; ++ct) {
    v8f acc; float bv = b1[ct * 16 + c];
    ACC_INIT(acc, bv);
#pragma unroll
    for (int j = 0; j < 4; ++j) {
      int k = 4 * j + 2 * kh;
      v2f b; b.x = sW1[k * 64 + ct * 16 + c];
             b.y = sW1[(k + 1) * 64 + ct * 16 + c];
      acc = wmma4(a1[j], b, acc);
    }
#pragma unroll
    for (int i = 0; i < 8; ++i)
      hid[wave][i + 8 * kh][ct * 16 + c] = leaky(acc[i]);
  }
  v2f a2[16];
#pragma unroll
  for (int j = 0; j < 16; ++j) {
    int k = 4 * j + 2 * kh;
    a2[j].x = hid[wave][r][k]; a2[j].y = hid[wave][r][k + 1];
  }
  for (int ct = 0; ct < 4; ++ct) {
    v8f acc; float bv = b2[ct * 16 + c];
    ACC_INIT(acc, bv);
#pragma unroll
    for (int j = 0; j < 16; ++j) {
      int k = 4 * j + 2 * kh;
      v2f b; b.x = sW2[k * 64 + ct * 16 + c];
             b.y = sW2[(k + 1) * 64 + ct * 16 + c];
      acc = wmma4(a2[j], b, acc);
    }
#pragma unroll
    for (int i = 0; i < 8; ++i)
      G[(size_t)(row0 + i + 8 * kh) * 64 + ct * 16 + c] = acc[i];
  }
}

// ---------------------------------------------------------------------------
// Elementwise / gather / scatter kernels
// ---------------------------------------------------------------------------
__global__ void bcast_vn_kernel(const float* __restrict__ emb, float* __restrict__ vn) {
  int t = blockIdx.x * blockDim.x + threadIdx.x;
  if (t < NUM_VN * EMB) vn[t] = emb[t & 63];
}

__global__ void fill_kernel(float* __restrict__ p, float v, int n) {
  int t = blockIdx.x * blockDim.x + threadIdx.x;
  if (t < n) p[t] = v;
}

__global__ void addvn_kernel(float* __restrict__ h, const float* __restrict__ vn,
                             const int* __restrict__ batch) {
  int t = blockIdx.x * blockDim.x + threadIdx.x;
  if (t >= N_NODES * EMB) return;
  int node = t >> 6, c = t & 63;
  h[t] += vn[(size_t)batch[node] * 64 + c];
}

// aggs[net] += h[sink_nodes[e]] * w[e]   (thread = edge x 4-col group)
__global__ void scatter_sink_kernel(const float* __restrict__ h,
                                    const int* __restrict__ sink_nodes,
                                    const int* __restrict__ sink_nets,
                                    const float* __restrict__ w,
                                    float* __restrict__ aggs) {
  int t = blockIdx.x * blockDim.x + threadIdx.x;
  if (t >= E_SINK * 16) return;
  int e = t >> 4, c4 = (t & 15) * 4;
  int node = sink_nodes[e], net = sink_nets[e];
  float ww = w[e];
  const float4 v = *(const float4*)(h + (size_t)node * 64 + c4);
  float* dst = aggs + (size_t)net * 64 + c4;
  atomicAdd(dst + 0, v.x * ww);
  atomicAdd(dst + 1, v.y * ww);
  atomicAdd(dst + 2, v.z * ww);
  atomicAdd(dst + 3, v.w * ww);
}

// back[src_nodes[net]] += g[net]
__global__ void scatter_src_back_kernel(const float* __restrict__ g,
                                        const int* __restrict__ src_nodes,
                                        float* __restrict__ back) {
  int t = blockIdx.x * blockDim.x + threadIdx.x;
  if (t >= N_NETS * 16) return;
  int net = t >> 4, c4 = (t & 15) * 4;
  const float4 v = *(const float4*)(g + (size_t)net * 64 + c4);
  float* dst = back + (size_t)src_nodes[net] * 64 + c4;
  atomicAdd(dst + 0, v.x);
  atomicAdd(dst + 1, v.y);
  atomicAdd(dst + 2, v.z);
  atomicAdd(dst + 3, v.w);
}

// back[sink_nodes[e]] += g[sink_nets[e]] * w[e]
__global__ void scatter_sink_back_kernel(const float* __restrict__ g,
                                         const int* __restrict__ sink_nodes,
                                         const int* __restrict__ sink_nets,
                                         const float* __restrict__ w,
                                         float* __restrict__ back) {
  int t = blockIdx.x * blockDim.x + threadIdx.x;
  if (t >= E_SINK * 16) return;
  int e = t >> 4, c4 = (t & 15) * 4;
  float ww = w[e];
  const float4 v = *(const float4*)(g + (size_t)sink_nets[e] * 64 + c4);
  float* dst = back + (size_t)sink_nodes[e] * 64 + c4;
  atomicAdd(dst + 0, v.x * ww);
  atomicAdd(dst + 1, v.y * ww);
  atomicAdd(dst + 2, v.z * ww);
  atomicAdd(dst + 3, v.w * ww);
}

__global__ void leaky_ip_kernel(float* __restrict__ p, int n) {
  int t = blockIdx.x * blockDim.x + threadIdx.x;
  if (t < n) p[t] = leaky(p[t]);
}

__global__ void pool_max_kernel(const float* __restrict__ h, const int* __restrict__ batch,
                                float* __restrict__ pooled) {
  int t = blockIdx.x * blockDim.x + threadIdx.x;
  if (t >= N_NODES * EMB) return;
  int node = t >> 6, c = t & 63;
  atomicMaxF(&pooled[(size_t)batch[node] * 64 + c], h[t]);
}

// ---------------------------------------------------------------------------
// Net conv: g_out = [g | h[src] | aggs] (16x192) @ W (192x64) + b  (pre-act)
// Concat is contiguous in K, so A frag j covers k = 4j+2kh.
// W staged in LDS (48KB). A frags (48 x v2f) preloaded once per wave.
// ---------------------------------------------------------------------------
__global__ void net_conv_kernel(const float* __restrict__ g, const float* __restrict__ h,
                                const float* __restrict__ aggs,
                                const int* __restrict__ src_nodes,
                                const float* __restrict__ W, const float* __restrict__ bias,
                                float* __restrict__ gout) {
  __shared__ __align__(16) float sW[192 * 64];
  int tid = threadIdx.x;
  int wave = tid >> 5, lane = tid & 31;
  int tile = blockIdx.x * 4 + wave;

  async_stage_issue(sW, W, 192 * 64, tid, 128);
  async_stage_wait();
  __syncthreads();
  if (tile >= N_NETS / 16) return;

  int row0 = tile * 16;
  int r = lane & 15, kh = lane >> 4, c = lane & 15;
  const float* grow = g + (size_t)(row0 + r) * 64;
  const float* arow = aggs + (size_t)(row0 + r) * 64;
  const float* srow = h + (size_t)src_nodes[row0 + r] * 64;

  v2f a[48];
#pragma unroll
  for (int j = 0; j < 48; ++j) {
    int k = 4 * j + 2 * kh;
    const float* src = (k < 64) ? (grow + k) : (k < 128) ? (srow + k - 64) : (arow + k - 128);
    a[j].x = src[0]; a[j].y = src[1];
  }
  for (int ct = 0; ct < 4; ++ct) {
    v8f acc; float bv = bias[ct * 16 + c];
    ACC_INIT(acc, bv);
#pragma unroll
    for (int j = 0; j < 48; ++j) {
      int k = 4 * j + 2 * kh;
      v2f b; b.x = sW[k * 64 + ct * 16 + c];
             b.y = sW[(k + 1) * 64 + ct * 16 + c];
      acc = wmma4(a[j], b, acc);
    }
#pragma unroll
    for (int i = 0; i < 8; ++i)
      gout[(size_t)(row0 + i + 8 * kh) * 64 + ct * 16 + c] = acc[i];
  }
}

// ---------------------------------------------------------------------------
// Node conv: h_out = leaky([h | back] (16x128) @ W (128x64) + b)
// ---------------------------------------------------------------------------
__global__ void node_conv_kernel(const float* __restrict__ h, const float* __restrict__ back,
                                 const float* __restrict__ W, const float* __restrict__ bias,
                                 float* __restrict__ hout) {
  __shared__ __align__(16) float sW[128 * 64];
  int tid = threadIdx.x;
  int wave = tid >> 5, lane = tid & 31;
  int tile = blockIdx.x * 4 + wave;

  async_stage_issue(sW, W, 128 * 64, tid, 128);
  async_stage_wait();
  __syncthreads();
  if (tile >= N_NODES / 16) return;

  int row0 = tile * 16;
  int r = lane & 15, kh = lane >> 4, c = lane & 15;
  const float* hrow = h + (size_t)(row0 + r) * 64;
  const float* brow = back + (size_t)(row0 + r) * 64;

  v2f a[32];
#pragma unroll
  for (int j = 0; j < 32; ++j) {
    int k = 4 * j + 2 * kh;
    const float* src = (k < 64) ? (hrow + k) : (brow + k - 64);
    a[j].x = src[0]; a[j].y = src[1];
  }
  for (int ct = 0; ct < 4; ++ct) {
    v8f acc; float bv = bias[ct * 16 + c];
    ACC_INIT(acc, bv);
#pragma unroll
    for (int j = 0; j < 32; ++j) {
      int k = 4 * j + 2 * kh;
      v2f b; b.x = sW[k * 64 + ct * 16 + c];
             b.y = sW[(k + 1) * 64 + ct * 16 + c];
      acc = wmma4(a[j], b, acc);
    }
#pragma unroll
    for (int i = 0; i < 8; ++i)
      hout[(size_t)(row0 + i + 8 * kh) * 64 + ct * 16 + c] = leaky(acc[i]);
  }
}

// ---------------------------------------------------------------------------
// Virtual-node MLP: vn_out = vn + leaky((pooled+vn)@W1+b1)@W2+b2  (1000 rows)
// ---------------------------------------------------------------------------
__global__ void vn_update_kernel(const float* __restrict__ pooled, const float* __restrict__ vn,
                                 const float* __restrict__ W1, const float* __restrict__ b1,
                                 const float* __restrict__ W2, const float* __restrict__ b2,
                                 float* __restrict__ vnout) {
  __shared__ __align__(16) float sW1[64 * 64];
  __shared__ __align__(16) float sW2[64 * 64];
  __shared__ __align__(16) float hid[4][16][66];
  int tid = threadIdx.x;
  int wave = tid >> 5, lane = tid & 31;
  int tile = blockIdx.x * 4 + wave;

  async_stage_issue(sW1, W1, 64 * 64, tid, 128);
  async_stage_issue(sW2, W2, 64 * 64, tid, 128);
  async_stage_wait();
  __syncthreads();
  const int NT = (NUM_VN + 15) / 16;  // 63
  if (tile >= NT) return;

  int row0 = tile * 16;
  int r = lane & 15, kh = lane >> 4, c = lane & 15;
  int row = row0 + r;
  bool valid = row < NUM_VN;

  v2f a1[16];
#pragma unroll
  for (int j = 0; j < 16; ++j) {
    int k = 4 * j + 2 * kh;
    a1[j].x = valid ? (pooled[(size_t)row * 64 + k] + vn[(size_t)row * 64 + k]) : 0.0f;
    a1[j].y = valid ? (pooled[(size_t)row * 64 + k + 1] + vn[(size_t)row * 64 + k + 1]) : 0.0f;
  }
  for (int ct = 0; ct < 4; ++ct) {
    v8f acc; float bv = b1[ct * 16 + c];
    ACC_INIT(acc, bv);
#pragma unroll
    for (int j = 0; j < 16; ++j) {
      int k = 4 * j + 2 * kh;
      v2f b; b.x = sW1[k * 64 + ct * 16 + c];
             b.y = sW1[(k + 1) * 64 + ct * 16 + c];
      acc = wmma4(a1[j], b, acc);
    }
#pragma unroll
    for (int i = 0; i < 8; ++i)
      hid[wave][i + 8 * kh][ct * 16 + c] = leaky(acc[i]);
  }
  v2f a2[16];
#pragma unroll
  for (int j = 0; j < 16; ++j) {
    int k = 4 * j + 2 * kh;
    a2[j].x = hid[wave][r][k]; a2[j].y = hid[wave][r][k + 1];
  }
  for (int ct = 0; ct < 4; ++ct) {
    v8f acc; float bv = b2[ct * 16 + c];
    ACC_INIT(acc, bv);
#pragma unroll
    for (int j = 0; j < 16; ++j) {
      int k = 4 * j + 2 * kh;
      v2f b; b.x = sW2[k * 64 + ct * 16 + c];
             b.y = sW2[(k + 1) * 64 + ct * 16 + c];
      acc = wmma4(a2[j], b, acc);
    }
#pragma unroll
    for (int i = 0; i < 8; ++i) {
      int ro = row0 + i + 8 * kh;
      if (ro < NUM_VN)
        vnout[(size_t)ro * 64 + ct * 16 + c] =
            vn[(size_t)ro * 64 + ct * 16 + c] + acc[i];
    }
  }
}

// ---------------------------------------------------------------------------
// Node head: out = |leaky(h@W1+b1)@W2+b2|, W1 64x256 (global), W2 256x4 (LDS).
// 2 waves/block; hidden 16x256 per wave in LDS.
// ---------------------------------------------------------------------------
__global__ void node_head_kernel(const float* __restrict__ h,
                                 const float* __restrict__ W1, const float* __restrict__ b1,
                                 const float* __restrict__ W2, const float* __restrict__ b2,
                                 float* __restrict__ out) {
  __shared__ __align__(16) float sW2[256 * 4];
  __shared__ __align__(16) float hid[2][16][258];
  int tid = threadIdx.x;
  int wave = tid >> 5, lane = tid & 31;
  int tile = blockIdx.x * 2 + wave;

  async_stage_issue(sW2, W2, 256 * 4, tid, 64);
  async_stage_wait();
  __syncthreads();
  if (tile >= N_NODES / 16) return;

  int row0 = tile * 16;
  int r = lane & 15, kh = lane >> 4, c = lane & 15;
  const float* hrow = h + (size_t)(row0 + r) * 64;

  v2f a1[16];
#pragma unroll
  for (int j = 0; j < 16; ++j) {
    int k = 4 * j + 2 * kh;
    a1[j].x = hrow[k]; a1[j].y = hrow[k + 1];
  }
  for (int ct = 0; ct < 16; ++ct) {             // 256 hidden cols
    v8f acc; float bv = b1[ct * 16 + c];
    ACC_INIT(acc, bv);
#pragma unroll
    for (int j = 0; j < 16; ++j) {
      int k = 4 * j + 2 * kh;
      v2f b; b.x = W1[(size_t)k * 256 + ct * 16 + c];
             b.y = W1[(size_t)(k + 1) * 256 + ct * 16 + c];
      acc = wmma4(a1[j], b, acc);
    }
#pragma unroll
    for (int i = 0; i < 8; ++i)
      hid[wave][i + 8 * kh][ct * 16 + c] = leaky(acc[i]);
  }
  {
    v8f acc; float bv = (c < 4) ? b2[c] : 0.0f;
    ACC_INIT(acc, bv);
#pragma unroll 8
    for (int j = 0; j < 64; ++j) {              // K = 256
      int k = 4 * j + 2 * kh;
      v2f a; a.x = hid[wave][r][k]; a.y = hid[wave][r][k + 1];
      v2f b;
      b.x = (c < 4) ? sW2[k * 4 + c] : 0.0f;
      b.y = (c < 4) ? sW2[(k + 1) * 4 + c] : 0.0f;
      acc = wmma4(a, b, acc);
    }
#pragma unroll
    for (int i = 0; i < 8; ++i)
      if (c < 4)
        out[(size_t)(row0 + i + 8 * kh) * 4 + c] = fabsf(acc[i]);
  }
}

// ---------------------------------------------------------------------------
// Net head: out = |leaky(g@W1+b1)@W2+b2|, W1 64x64 + W2 64x4 staged in LDS.
// ---------------------------------------------------------------------------
__global__ void net_head_kernel(const float* __restrict__ g,
                                const float* __restrict__ W1, const float* __restrict__ b1,
                                const float* __restrict__ W2, const float* __restrict__ b2,
                                float* __restrict__ out) {
  __shared__ __align__(16) float sW1[64 * 64];
  __shared__ __align__(16) float sW2[64 * 4];
  __shared__ __align__(16) float hid[4][16][66];
  int tid = threadIdx.x;
  int wave = tid >> 5, lane = tid & 31;
  int tile = blockIdx.x * 4 + wave;

  async_stage_issue(sW1, W1, 64 * 64, tid, 128);
  async_stage_issue(sW2, W2, 64 * 4, tid, 128);
  async_stage_wait();
  __syncthreads();
  if (tile >= N_NETS / 16) return;

  int row0 = tile * 16;
  int r = lane & 15, kh = lane >> 4, c = lane & 15;
  const float* grow = g + (size_t)(row0 + r) * 64;

  v2f a1[16];
#pragma unroll
  for (int j = 0; j < 16; ++j) {
    int k = 4 * j + 2 * kh;
    a1[j].x = grow[k]; a1[j].y = grow[k + 1];
  }
  for (int ct = 0; ct < 4; ++ct) {
    v8f acc; float bv = b1[ct * 16 + c];
    ACC_INIT(acc, bv);
#pragma unroll
    for (int j = 0; j < 16; ++j) {
      int k = 4 * j + 2 * kh;
      v2f b; b.x = sW1[k * 64 + ct * 16 + c];
             b.y = sW1[(k + 1) * 64 + ct * 16 + c];
      acc = wmma4(a1[j], b, acc);
    }
#pragma unroll
    for (int i = 0; i < 8; ++i)
      hid[wave][i + 8 * kh][ct * 16 + c] = leaky(acc[i]);
  }
  {
    v8f acc; float bv = (c < 4) ? b2[c] : 0.0f;
    ACC_INIT(acc, bv);
#pragma unroll
    for (int j = 0; j < 16; ++j) {
      int k = 4 * j + 2 * kh;
      v2f a; a.x = hid[wave][r][k]; a.y = hid[wave][r][k + 1];
      v2f b;
      b.x = (c < 4) ? sW2[k * 4 + c] : 0.0f;
      b.y = (c < 4) ? sW2[(k + 1) * 4 + c] : 0.0f;
      acc = wmma4(a, b, acc);
    }
#pragma unroll
    for (int i = 0; i < 8; ++i)
      if (c < 4)
        out[(size_t)(row0 + i + 8 * kh) * 4 + c] = fabsf(acc[i]);
  }
}

// ---------------------------------------------------------------------------
// Host driver
// ---------------------------------------------------------------------------
extern "C" void kernel_launch(void* const* d_in, const int* in_sizes, int n_in,
                              void* d_out, int out_size, void* d_ws, size_t ws_size,
                              hipStream_t stream) {
  (void)in_sizes; (void)n_in; (void)out_size; (void)ws_size;

  const float* node_features = (const float*)d_in[0];
  const float* net_features  = (const float*)d_in[1];
  const int*   src_nodes     = (const int*)d_in[2];
  const int*   sink_nodes    = (const int*)d_in[3];
  const int*   sink_nets     = (const int*)d_in[4];
  const float* edge_weight   = (const float*)d_in[5];
  const int*   batch         = (const int*)d_in[6];
  const float* ne_W1 = (const float*)d_in[7];
  const float* ne_b1 = (const float*)d_in[8];
  const float* ne_W2 = (const float*)d_in[9];
  const float* ne_b2 = (const float*)d_in[10];
  const float* te_W1 = (const float*)d_in[11];
  const float* te_b1 = (const float*)d_in[12];
  const float* te_W2 = (const float*)d_in[13];
  const float* te_b2 = (const float*)d_in[14];
  const float* vn_emb     = (const float*)d_in[15];
  const float* conv_Wnet  = (const float*)d_in[16];
  const float* conv_bnet  = (const float*)d_in[17];
  const float* conv_Wnode = (const float*)d_in[18];
  const float* conv_bnode = (const float*)d_in[19];
  const float* vn_W1 = (const float*)d_in[20];
  const float* vn_b1 = (const float*)d_in[21];
  const float* vn_W2 = (const float*)d_in[22];
  const float* vn_b2 = (const float*)d_in[23];
  const float* fc1n_W = (const float*)d_in[24];
  const float* fc1n_b = (const float*)d_in[25];
  const float* fc2n_W = (const float*)d_in[26];
  const float* fc2n_b = (const float*)d_in[27];
  const float* fc1e_W = (const float*)d_in[28];
  const float* fc1e_b = (const float*)d_in[29];
  const float* fc2e_W = (const float*)d_in[30];
  const float* fc2e_b = (const float*)d_in[31];

  // Workspace carve-out
  char* p = (char*)d_ws;
  auto alloc = [&](size_t bytes) -> float* {
    float* q = (float*)p;
    p += (bytes + 255) & ~(size_t)255;
    return q;
  };
  const size_t HB = (size_t)N_NODES * 64 * sizeof(float);   // 76.8 MB
  const size_t GB = (size_t)N_NETS  * 64 * sizeof(float);   // 51.2 MB
  const size_t VB = (size_t)NUM_VN  * 64 * sizeof(float);
  float* hA   = alloc(HB);
  float* hBuf = alloc(HB);
  float* gA   = alloc(GB);
  float* gBuf = alloc(GB);
  float* aggs = alloc(GB);
  float* back = alloc(HB);
  float* vnA  = alloc(VB);
  float* vnB  = alloc(VB);
  float* pooled = alloc(VB);

  float* out_node = (float*)d_out;
  float* out_net  = out_node + (size_t)N_NODES * 4;

  float* h = hA; float* h2 = hBuf;
  float* g = gA; float* g2 = gBuf;
  float* vn = vnA; float* vn2 = vnB;

  const int NODE_TILES = N_NODES / 16;  // 18750
  const int NET_TILES  = N_NETS / 16;   // 12500
  const int VN_TILES   = (NUM_VN + 15) / 16;  // 63

  // Encoders
  enc_node_kernel<<<(NODE_TILES + 1) / 2, 64, 0, stream>>>(
      node_features, ne_W1, ne_b1, ne_W2, ne_b2, h);
  enc_net_kernel<<<(NET_TILES + 3) / 4, 128, 0, stream>>>(
      net_features, te_W1, te_b1, te_W2, te_b2, g);
  bcast_vn_kernel<<<(NUM_VN * EMB + 255) / 256, 256, 0, stream>>>(vn_emb, vn);

  for (int l = 0; l < 3; ++l) {
    const float* Wnet  = conv_Wnet  + (size_t)l * 192 * 64;
    const float* bnet  = conv_bnet  + (size_t)l * 64;
    const float* Wnode = conv_Wnode + (size_t)l * 128 * 64;
    const float* bnode = conv_bnode + (size_t)l * 64;

    // h += vn[batch]
    addvn_kernel<<<(N_NODES * 64 + 255) / 256, 256, 0, stream>>>(h, vn, batch);

    // agg_sink = segsum(h[sink_nodes]*w, sink_nets)
    hipMemsetAsync(aggs, 0, GB, stream);
    scatter_sink_kernel<<<(E_SINK * 16 + 255) / 256, 256, 0, stream>>>(
        h, sink_nodes, sink_nets, edge_weight, aggs);

    // g2 = [g | h[src] | agg_sink] @ Wnet + bnet   (pre-activation)
    net_conv_kernel<<<(NET_TILES + 3) / 4, 128, 0, stream>>>(
        g, h, aggs, src_nodes, Wnet, bnet, g2);

    // back = segsum(g2, src_nodes) + segsum(g2[sink_nets]*w, sink_nodes)
    hipMemsetAsync(back, 0, HB, stream);
    scatter_src_back_kernel<<<(N_NETS * 16 + 255) / 256, 256, 0, stream>>>(
        g2, src_nodes, back);
    scatter_sink_back_kernel<<<(E_SINK * 16 + 255) / 256, 256, 0, stream>>>(
        g2, sink_nodes, sink_nets, edge_weight, back);

    // h2 = leaky([h | back] @ Wnode + bnode)
    node_conv_kernel<<<(NODE_TILES + 3) / 4, 128, 0, stream>>>(
        h, back, Wnode, bnode, h2);

    // g2 = leaky(g2)
    leaky_ip_kernel<<<(N_NETS * 64 + 255) / 256, 256, 0, stream>>>(g2, N_NETS * 64);

    if (l < 2) {
      fill_kernel<<<(NUM_VN * EMB + 255) / 256, 256, 0, stream>>>(
          pooled, -FLT_MAX, NUM_VN * EMB);
      pool_max_kernel<<<(N_NODES * 64 + 255) / 256, 256, 0, stream>>>(h2, batch, pooled);
      vn_update_kernel<<<(VN_TILES + 3) / 4, 128, 0, stream>>>(
          pooled, vn,
          vn_W1 + (size_t)l * 64 * 64, vn_b1 + (size_t)l * 64,
          vn_W2 + (size_t)l * 64 * 64, vn_b2 + (size_t)l * 64, vn2);
      float* tv = vn; vn = vn2; vn2 = tv;
    }
    float* th = h; h = h2; h2 = th;
    float* tg = g; g = g2; g2 = tg;
  }

  // Output heads
  node_head_kernel<<<(NODE_TILES + 1) / 2, 64, 0, stream>>>(
      h, fc1n_W, fc1n_b, fc2n_W, fc2n_b, out_node);
  net_head_kernel<<<(NET_TILES + 3) / 4, 128, 0, stream>>>(
      g, fc1e_W, fc1e_b, fc2e_W, fc2e_b, out_net);
}